// MultiboxLoss_26499948216438
// MI455X (gfx1250) — compile-verified
//
#include <hip/hip_runtime.h>
#include <stdint.h>
#include <stddef.h>

#define B_    64
#define P_    8732
#define C_    81
#define TILE  128
#define NTIL  ((P_ + TILE - 1) / TILE)   /* 69 */
#define NBIN  4096
#define LCAP  2048

// ---------------- CDNA5 async global->LDS path (guarded) ----------------
#if defined(__HIP_DEVICE_COMPILE__) && \
    __has_builtin(__builtin_amdgcn_global_load_async_to_lds_b128)
#define ASYNC_LDS 1
typedef int v4i __attribute__((ext_vector_type(4)));
typedef __attribute__((address_space(1))) v4i GV4;   // global int4
typedef __attribute__((address_space(3))) v4i LV4;   // LDS int4
__device__ __forceinline__ void lds_async16(const void* g, void* l) {
  __builtin_amdgcn_global_load_async_to_lds_b128((GV4*)g, (LV4*)l, 0, 0);
}
#if __has_builtin(__builtin_amdgcn_s_wait_asynccnt)
#define WAIT_ASYNC() __builtin_amdgcn_s_wait_asynccnt(0)
#else
#define WAIT_ASYNC() asm volatile("s_wait_asynccnt 0" ::: "memory")
#endif
#else
#define WAIT_ASYNC() ((void)0)
#endif

// ---------------- Kernel 1: log-sum-exp stream + per-tile partials ------
__global__ __launch_bounds__(TILE) void k_lse(
    const float* __restrict__ conf, const int* __restrict__ labels,
    const float* __restrict__ ploc, const float* __restrict__ gloc,
    float* __restrict__ negprob, float* __restrict__ nll,
    int* __restrict__ posPart, float* __restrict__ sl1Part)
{
  __shared__ float lconf[TILE * C_];   // 41472 B, packed 324B rows (0-conflict)
  __shared__ float rs[TILE];
  __shared__ int   ri[TILE];
  const int t   = threadIdx.x;
  const int bid = blockIdx.x;
  const int b   = bid / NTIL;
  const int tp  = bid - b * NTIL;
  const int p0  = tp * TILE;
  const int nv  = min(TILE, P_ - p0);

  // Stage the tile of confidence rows into LDS as one linear block.
  // Base element index (b*P + p0)*81 is a multiple of 4 -> 16B aligned,
  // and nv is always a multiple of 4 -> byte count is a multiple of 16.
  {
    const char* g  = (const char*)(conf + ((size_t)b * P_ + p0) * (size_t)C_);
    char*       l  = (char*)lconf;
    const int  n16 = (nv * C_ * 4) >> 4;
#ifdef ASYNC_LDS
    for (int i = t; i < n16; i += TILE) lds_async16(g + 16 * i, l + 16 * i);
    WAIT_ASYNC();
#else
    for (int i = t; i < n16; i += TILE)
      *(float4*)(l + 16 * i) = *(const float4*)(g + 16 * i);
#endif
  }
  __syncthreads();

  int   pos = 0;
  float sl1 = 0.0f;
  if (t < nv) {
    const float* r = lconf + t * C_;
    float m = r[0];
#pragma unroll 4
    for (int c = 1; c < C_; ++c) m = fmaxf(m, r[c]);
    float s = 0.0f;
#pragma unroll 4
    for (int c = 0; c < C_; ++c) s += __expf(r[c] - m);
    const float  lse = m + __logf(s);
    const size_t gp  = (size_t)b * P_ + p0 + t;
    const int    lab = labels[gp];
    pos = (lab > 0);
    negprob[gp] = pos ? -1.0f : (lse - r[0]);   // -logp[0], positives excluded
    nll[gp]     = lse - r[lab];                 // -logp[label]
    if (pos) {
      const float4 pd = *(const float4*)(ploc + gp * 4);
      const float4 gd = *(const float4*)(gloc + gp * 4);
      float d, a;
      d = pd.x - gd.x; a = fabsf(d); sl1 += (a < 1.0f) ? 0.5f * d * d : a - 0.5f;
      d = pd.y - gd.y; a = fabsf(d); sl1 += (a < 1.0f) ? 0.5f * d * d : a - 0.5f;
      d = pd.z - gd.z; a = fabsf(d); sl1 += (a < 1.0f) ? 0.5f * d * d : a - 0.5f;
      d = pd.w - gd.w; a = fabsf(d); sl1 += (a < 1.0f) ? 0.5f * d * d : a - 0.5f;
    }
  }
  ri[t] = pos; rs[t] = sl1;
  __syncthreads();
  for (int s2 = TILE / 2; s2; s2 >>= 1) {
    if (t < s2) { ri[t] += ri[t + s2]; rs[t] += rs[t + s2]; }
    __syncthreads();
  }
  if (t == 0) { posPart[bid] = ri[0]; sl1Part[bid] = rs[0]; }
}

// ---------------- Kernel 2: exact top-K hard-negative mining per row ----
__global__ __launch_bounds__(1024) void k_select(
    const float* __restrict__ negprob, const float* __restrict__ nll,
    const int* __restrict__ posPart,
    int* __restrict__ numpos, float* __restrict__ nllRow, int* __restrict__ selRow)
{
  __shared__ int   hist[NBIN];
  __shared__ float lv[LCAP];
  __shared__ int   li[LCAP];
  __shared__ int   nlist, sT, sR, s_np;
  __shared__ float rf[1024];
  __shared__ int   rn[1024];
  const int b = blockIdx.x, t = threadIdx.x;
  const float* v  = negprob + (size_t)b * P_;
  const float* nl = nll     + (size_t)b * P_;

  for (int i = t; i < NBIN; i += 1024) hist[i] = 0;
  if (t == 0) {
    nlist = 0;
    int s = 0;
    for (int i = 0; i < NTIL; ++i) s += posPart[b * NTIL + i];
    s_np = s;
    numpos[b] = s;
  }
  __syncthreads();

  // Histogram over float bit patterns (monotone for x >= 0).
  for (int p = t; p < P_; p += 1024) {
    const float x = v[p];
    if (x >= 0.0f)
      atomicAdd(&hist[(__float_as_uint(x) >> 19) & (NBIN - 1)], 1);
  }
  __syncthreads();

  // Suffix counts (cnt_ge) and threshold bin for K = min(3*num_pos, #neg).
  if (t == 0) {
    int acc = 0;
    for (int k = NBIN - 1; k >= 0; --k) { acc += hist[k]; hist[k] = acc; }
    const int negc = hist[0];
    int K = 3 * s_np; if (K > negc) K = negc;
    if (K <= 0) { sT = NBIN; sR = 0; }
    else {
      int T = 0;
      for (int k = 1; k < NBIN; ++k) { if (hist[k] >= K) T = k; else break; }
      const int gt = (T + 1 < NBIN) ? hist[T + 1] : 0;
      sT = T; sR = K - gt;      // take R items from threshold bin T
    }
  }
  __syncthreads();
  const int T = sT, R = sR;

  // Gather items falling in the threshold bin (typically a handful).
  if (T < NBIN) {
    for (int p = t; p < P_; p += 1024) {
      const float x = v[p];
      if (x >= 0.0f && (int)((__float_as_uint(x) >> 19) & (NBIN - 1)) == T) {
        const int idx = atomicAdd(&nlist, 1);
        if (idx < LCAP) { lv[idx] = x; li[idx] = p; }
      }
    }
  }
  __syncthreads();
  const int L = min(nlist, LCAP);

  // Selection + masked-NLL accumulation (exact stable-sort tie-break).
  float accn = 0.0f; int accc = 0;
  for (int p = t; p < P_; p += 1024) {
    const float x = v[p];
    bool sel;
    if (x < 0.0f) sel = true;                 // positive prior
    else {
      const int k = (int)((__float_as_uint(x) >> 19) & (NBIN - 1));
      if (k > T) sel = true;
      else if (k == T) {
        int r = 0;
        for (int i = 0; i < L; ++i)
          r += (int)((lv[i] > x) | ((lv[i] == x) & (li[i] < p)));
        sel = (r < R);
      } else sel = false;
    }
    if (sel) { accn += nl[p]; ++accc; }
  }
  rf[t] = accn; rn[t] = accc;
  __syncthreads();
  for (int s2 = 512; s2; s2 >>= 1) {
    if (t < s2) { rf[t] += rf[t + s2]; rn[t] += rn[t + s2]; }
    __syncthreads();
  }
  if (t == 0) { nllRow[b] = rf[0]; selRow[b] = rn[0]; }
}

// ---------------- Kernel 3: deterministic finalize ----------------------
__global__ __launch_bounds__(256) void k_final(
    const int* __restrict__ posPart, const float* __restrict__ sl1Part,
    const float* __restrict__ nllRow, const int* __restrict__ selRow,
    const int* __restrict__ numpos, float* __restrict__ out)
{
  __shared__ float rf[256];
  __shared__ int   rn[256];
  __shared__ float s_sl1, s_ce;
  __shared__ int   s_tp;
  const int t = threadIdx.x;
  float fs = 0.0f; int is = 0;
  for (int i = t; i < B_ * NTIL; i += 256) { fs += sl1Part[i]; is += posPart[i]; }
  rf[t] = fs; rn[t] = is;
  __syncthreads();
  for (int s2 = 128; s2; s2 >>= 1) {
    if (t < s2) { rf[t] += rf[t + s2]; rn[t] += rn[t + s2]; }
    __syncthreads();
  }
  if (t == 0) {
    s_sl1 = rf[0]; s_tp = rn[0];
    float ns = 0.0f; int sc = 0;
    for (int i = 0; i < B_; ++i) { ns += nllRow[i]; sc += selRow[i]; }
    s_ce = ns / fmaxf((float)sc, 1.0f);
  }
  __syncthreads();
  if (t < B_) out[t] = s_ce / (float)numpos[t];            // conf_loss (B,1)
  if (t == 0) out[B_] = s_sl1 / fmaxf((float)s_tp, 1.0f);  // loc_huber_loss
}

// ---------------- Host launcher -----------------------------------------
extern "C" void kernel_launch(void* const* d_in, const int* in_sizes, int n_in,
                              void* d_out, int out_size, void* d_ws, size_t ws_size,
                              hipStream_t stream) {
  (void)in_sizes; (void)n_in; (void)out_size; (void)ws_size;
  const float* conf   = (const float*)d_in[0];
  const float* ploc   = (const float*)d_in[1];
  const int*   labels = (const int*)d_in[2];
  const float* gloc   = (const float*)d_in[3];
  float* out = (float*)d_out;

  char* w = (char*)d_ws;
  float* negprob = (float*)w; w += sizeof(float) * (size_t)B_ * P_;
  float* nll     = (float*)w; w += sizeof(float) * (size_t)B_ * P_;
  int*   posPart = (int*)w;   w += sizeof(int)   * (size_t)B_ * NTIL;
  float* sl1Part = (float*)w; w += sizeof(float) * (size_t)B_ * NTIL;
  int*   numpos  = (int*)w;   w += sizeof(int)   * B_;
  float* nllRow  = (float*)w; w += sizeof(float) * B_;
  int*   selRow  = (int*)w;   w += sizeof(int)   * B_;

  k_lse<<<B_ * NTIL, TILE, 0, stream>>>(conf, labels, ploc, gloc,
                                        negprob, nll, posPart, sl1Part);
  k_select<<<B_, 1024, 0, stream>>>(negprob, nll, posPart, numpos, nllRow, selRow);
  k_final<<<1, 256, 0, stream>>>(posPart, sl1Part, nllRow, selRow, numpos, out);
}